// MaxwellTM_79267916415435
// MI455X (gfx1250) — compile-verified
//
#include <hip/hip_runtime.h>
#include <stdint.h>

// ---------------------------------------------------------------------------
// FDTD-TM + CPML single step, fp32, bandwidth-bound (~620 MB compulsory
// traffic -> ~27us floor @ 23.3 TB/s HBM). Two kernels; the new-Ey
// intermediate (33 MB) stays resident in the 192 MB L2 between them.
// CDNA5-specific paths: GLOBAL_LOAD_ASYNC_TO_LDS_B128/B32 (ASYNCcnt) for
// x-halo staging, S_WAIT_ASYNCCNT, GLOBAL_PREFETCH_B8 for y-halo rows,
// wave32 256-thread row tiles.
// ---------------------------------------------------------------------------

#define TILE 256
#define HALO 8   // covers up to stencil order 8 (4 taps/side), keeps 16B align
#define SPAN (TILE + 2 * HALO)   // 272 floats = 68 x 16B

__device__ __forceinline__ void cdna5_async_load_f32(float* lds_dst,
                                                     const float* gsrc) {
#if defined(__gfx1250__)
  // Generic pointer to __shared__ carries the wave-relative LDS offset in its
  // low 32 bits (LDS aperture encoding) -> that is the VDST operand.
  uint32_t lds_off = (uint32_t)(uintptr_t)lds_dst;
  asm volatile("global_load_async_to_lds_b32 %0, %1, off"
               :: "v"(lds_off), "v"(gsrc)
               : "memory");
#else
  *lds_dst = *gsrc;
#endif
}

__device__ __forceinline__ void cdna5_async_load_f32x4(float* lds_dst,
                                                       const float* gsrc) {
#if defined(__gfx1250__)
  uint32_t lds_off = (uint32_t)(uintptr_t)lds_dst;
  asm volatile("global_load_async_to_lds_b128 %0, %1, off"
               :: "v"(lds_off), "v"(gsrc)
               : "memory");
#else
  lds_dst[0] = gsrc[0]; lds_dst[1] = gsrc[1];
  lds_dst[2] = gsrc[2]; lds_dst[3] = gsrc[3];
#endif
}

__device__ __forceinline__ void cdna5_wait_async() {
#if defined(__gfx1250__)
  asm volatile("s_wait_asynccnt 0" ::: "memory");
#endif
}

__device__ __forceinline__ void cdna5_prefetch(const void* p) {
#if defined(__gfx1250__)
  __builtin_prefetch(p, 0, 3);   // -> global_prefetch_b8
#endif
}

// Stage one row segment [x0-HALO, x0+TILE+HALO) into LDS with zero padding.
__device__ __forceinline__ void stage_row(float* s, const float* row,
                                          int x0, int NX, int tid) {
  const int g0 = x0 - HALO;
  if (g0 >= 0 && g0 + SPAN <= NX && (NX & 3) == 0) {
    // Fully interior + 16B aligned: wide async path (68 lanes x 16B).
    if (tid < SPAN / 4) cdna5_async_load_f32x4(&s[4 * tid], row + g0 + 4 * tid);
  } else {
    for (int i = tid; i < SPAN; i += TILE) {
      int gx = g0 + i;
      if (gx >= 0 && gx < NX) cdna5_async_load_f32(&s[i], row + gx);
      else                    s[i] = 0.0f;      // zero-padded boundary
    }
  }
}

// Taylor-optimal staggered-grid coefficients, row = stencil/2 - 1
__device__ __constant__ float kCoef[4][4] = {
  {1.0f,               0.0f,               0.0f,              0.0f},
  {9.0f/8.0f,         -1.0f/24.0f,         0.0f,              0.0f},
  {75.0f/64.0f,       -25.0f/384.0f,       3.0f/640.0f,       0.0f},
  {1225.0f/1024.0f,   -245.0f/3072.0f,     49.0f/5120.0f,    -5.0f/7168.0f}};

// ---------------------------------------------------------------------------
// Kernel 1: update_E.  Writes Ey(new), m_Hx_z(new), m_Hz_x(new).
// ---------------------------------------------------------------------------
__global__ void __launch_bounds__(TILE)
fdtd_update_E(const float* __restrict__ ca,  const float* __restrict__ cb,
              const float* __restrict__ Ey,  const float* __restrict__ Hx,
              const float* __restrict__ Hz,
              const float* __restrict__ mHxz_in, const float* __restrict__ mHzx_in,
              const float* __restrict__ ky, const float* __restrict__ ay,
              const float* __restrict__ by,
              const float* __restrict__ kx, const float* __restrict__ ax,
              const float* __restrict__ bx,
              const float* __restrict__ rdy_p, const float* __restrict__ rdx_p,
              const int*   __restrict__ stencil_p,
              float* __restrict__ outEy, float* __restrict__ outMHxz,
              float* __restrict__ outMHzx,
              int NY, int NX)
{
  const int  tid  = threadIdx.x;
  const int  x0   = blockIdx.x * TILE;
  const int  y    = blockIdx.y;
  const int  b    = blockIdx.z;
  const long plane   = (long)NY * (long)NX;
  const long rowbase = (long)b * plane + (long)y * (long)NX;
  const int  x    = x0 + tid;

  // ---- stage Hz row (with x halo) into LDS via async loads ----
  __shared__ float sHz[SPAN];
  stage_row(sHz, Hz + rowbase, x0, NX, tid);

  // Overlap: prefetch the y-halo Hx rows while the async staging is in flight.
  if (x < NX) {
    if (y >= 1)      cdna5_prefetch(&Hx[rowbase - 1 * NX + x]);
    if (y + 1 < NY)  cdna5_prefetch(&Hx[rowbase + 1 * NX + x]);
  }

  cdna5_wait_async();
  __syncthreads();

  if (x >= NX) return;

  const int    N   = stencil_p[0] >> 1;           // taps per side (1..4)
  const float* C   = kCoef[N - 1];
  const float  rdx = rdx_p[0];
  const float  rdy = rdy_p[0];

  // d(Hz)/dx  (integer-grid):  sum c_k (a[i+k-1] - a[i-k]) * rdx
  const int ic = tid + HALO;
  float dHz_dx = 0.0f;
#pragma unroll
  for (int k = 1; k <= 4; ++k) {
    if (k > N) break;
    dHz_dx += C[k - 1] * (sHz[ic + (k - 1)] - sHz[ic - k]);
  }
  dHz_dx *= rdx;

  // d(Hx)/dz (integer-grid along y): coalesced row loads, L2 gives 4x reuse
  float dHx_dz = 0.0f;
#pragma unroll
  for (int k = 1; k <= 4; ++k) {
    if (k > N) break;
    const int yp = y + (k - 1);
    const int ym = y - k;
    float hp = (yp < NY) ? Hx[(long)b * plane + (long)yp * NX + x] : 0.0f;
    float hm = (ym >= 0) ? Hx[(long)b * plane + (long)ym * NX + x] : 0.0f;
    dHx_dz += C[k - 1] * (hp - hm);
  }
  dHx_dz *= rdy;

  const long idx  = rowbase + x;
  const long cidx = (long)y * NX + x;

  const float mzx = bx[x] * mHzx_in[idx] + ax[x] * dHz_dx;
  const float mxz = by[y] * mHxz_in[idx] + ay[y] * dHx_dz;
  const float ey  = ca[cidx] * Ey[idx] +
                    cb[cidx] * ((dHz_dx / kx[x] + mzx) - (dHx_dz / ky[y] + mxz));

  outEy[idx]   = ey;
  outMHzx[idx] = mzx;
  outMHxz[idx] = mxz;
}

// ---------------------------------------------------------------------------
// Kernel 2: update_H from the NEW Ey (L2-resident). Writes Hx, Hz, m_Ey_x, m_Ey_z.
// ---------------------------------------------------------------------------
__global__ void __launch_bounds__(TILE)
fdtd_update_H(const float* __restrict__ cq,
              const float* __restrict__ EyN,          // new Ey (d_out plane 0)
              const float* __restrict__ Hx, const float* __restrict__ Hz,
              const float* __restrict__ mEyx_in, const float* __restrict__ mEyz_in,
              const float* __restrict__ kyh, const float* __restrict__ ayh,
              const float* __restrict__ byh,
              const float* __restrict__ kxh, const float* __restrict__ axh,
              const float* __restrict__ bxh,
              const float* __restrict__ rdy_p, const float* __restrict__ rdx_p,
              const int*   __restrict__ stencil_p,
              float* __restrict__ outHx,  float* __restrict__ outHz,
              float* __restrict__ outMEyx, float* __restrict__ outMEyz,
              int NY, int NX)
{
  const int  tid  = threadIdx.x;
  const int  x0   = blockIdx.x * TILE;
  const int  y    = blockIdx.y;
  const int  b    = blockIdx.z;
  const long plane   = (long)NY * (long)NX;
  const long rowbase = (long)b * plane + (long)y * (long)NX;
  const int  x    = x0 + tid;

  // ---- stage new-Ey row (with x halo) into LDS via async loads ----
  __shared__ float sEy[SPAN];
  stage_row(sEy, EyN + rowbase, x0, NX, tid);

  if (x < NX) {
    if (y + 1 < NY)  cdna5_prefetch(&EyN[rowbase + 1 * NX + x]);
    if (y >= 1)      cdna5_prefetch(&EyN[rowbase - 1 * NX + x]);
  }

  cdna5_wait_async();
  __syncthreads();

  if (x >= NX) return;

  const int    N   = stencil_p[0] >> 1;
  const float* C   = kCoef[N - 1];
  const float  rdx = rdx_p[0];
  const float  rdy = rdy_p[0];

  // d(Ey)/dx (half-grid):  sum c_k (a[i+k] - a[i-k+1]) * rdx
  const int ic = tid + HALO;
  float dEy_dx = 0.0f;
#pragma unroll
  for (int k = 1; k <= 4; ++k) {
    if (k > N) break;
    dEy_dx += C[k - 1] * (sEy[ic + k] - sEy[ic - (k - 1)]);
  }
  dEy_dx *= rdx;

  // d(Ey)/dz (half-grid along y), reads hit L2 (written by kernel 1)
  float dEy_dz = 0.0f;
#pragma unroll
  for (int k = 1; k <= 4; ++k) {
    if (k > N) break;
    const int yp = y + k;
    const int ym = y - (k - 1);
    float ep = (yp < NY) ? EyN[(long)b * plane + (long)yp * NX + x] : 0.0f;
    float em = (ym >= 0) ? EyN[(long)b * plane + (long)ym * NX + x] : 0.0f;
    dEy_dz += C[k - 1] * (ep - em);
  }
  dEy_dz *= rdy;

  const long idx  = rowbase + x;
  const long cidx = (long)y * NX + x;

  const float mez = byh[y] * mEyz_in[idx] + ayh[y] * dEy_dz;
  const float mex = bxh[x] * mEyx_in[idx] + axh[x] * dEy_dx;

  outHx[idx]   = Hx[idx] - cq[cidx] * (dEy_dz / kyh[y] + mez);
  outHz[idx]   = Hz[idx] + cq[cidx] * (dEy_dx / kxh[x] + mex);
  outMEyz[idx] = mez;
  outMEyx[idx] = mex;
}

// ---------------------------------------------------------------------------
extern "C" void kernel_launch(void* const* d_in, const int* in_sizes, int n_in,
                              void* d_out, int out_size, void* d_ws, size_t ws_size,
                              hipStream_t stream) {
  // setup_inputs() order:
  const float* ca    = (const float*)d_in[0];
  const float* cb    = (const float*)d_in[1];
  const float* cq    = (const float*)d_in[2];
  const float* Ey    = (const float*)d_in[3];
  const float* Hx    = (const float*)d_in[4];
  const float* Hz    = (const float*)d_in[5];
  const float* mHxz  = (const float*)d_in[6];
  const float* mHzx  = (const float*)d_in[7];
  const float* mEyx  = (const float*)d_in[8];
  const float* mEyz  = (const float*)d_in[9];
  const float* ky    = (const float*)d_in[10];
  const float* kyh   = (const float*)d_in[11];
  const float* ay    = (const float*)d_in[12];
  const float* ayh   = (const float*)d_in[13];
  const float* by    = (const float*)d_in[14];
  const float* byh   = (const float*)d_in[15];
  const float* kx    = (const float*)d_in[16];
  const float* kxh   = (const float*)d_in[17];
  const float* ax    = (const float*)d_in[18];
  const float* axh   = (const float*)d_in[19];
  const float* bx    = (const float*)d_in[20];
  const float* bxh   = (const float*)d_in[21];
  const float* rdy   = (const float*)d_in[22];
  const float* rdx   = (const float*)d_in[23];
  const int*   stenc = (const int*)d_in[24];

  const int  NY   = in_sizes[10];             // ky has NY elements
  const int  NX   = in_sizes[16];             // kx has NX elements
  const long NYNX = (long)in_sizes[0];        // ca is (NY, NX)
  const int  B    = (int)((long)in_sizes[3] / NYNX);

  float* out = (float*)d_out;
  const long P = (long)B * NY * NX;           // one stacked output plane
  float* oEy   = out + 0 * P;
  float* oHx   = out + 1 * P;
  float* oHz   = out + 2 * P;
  float* oMHxz = out + 3 * P;
  float* oMHzx = out + 4 * P;
  float* oMEyx = out + 5 * P;
  float* oMEyz = out + 6 * P;

  dim3 grid((NX + TILE - 1) / TILE, NY, B);

  fdtd_update_E<<<grid, TILE, 0, stream>>>(
      ca, cb, Ey, Hx, Hz, mHxz, mHzx,
      ky, ay, by, kx, ax, bx, rdy, rdx, stenc,
      oEy, oMHxz, oMHzx, NY, NX);

  fdtd_update_H<<<grid, TILE, 0, stream>>>(
      cq, oEy, Hx, Hz, mEyx, mEyz,
      kyh, ayh, byh, kxh, axh, bxh, rdy, rdx, stenc,
      oHx, oHz, oMEyx, oMEyz, NY, NX);
}